// BalanceBCELoss_75024488727218
// MI455X (gfx1250) — compile-verified
//
#include <hip/hip_runtime.h>
#include <stdint.h>

// ---------------------------------------------------------------------------
// Balance BCE loss with hard-negative mining, exact top-k via 3-level radix
// select on float bit patterns (11+11+10 bits). All reductions use integer /
// fixed-point atomics -> bit-deterministic across replays.
//
// Block-local histograms pack {count, fixed-point sum} into a single u64 so
// each element needs only ONE ds_add_u64 (count in bits 63:50, sum in 49:0).
// Bounds: <=~8k elements/block (<2^14), sum <= 8k*100*2^23 ~ 6.7e12 < 2^50.
// ---------------------------------------------------------------------------

#define FSCALE    8388608.0f   // 2^23 fixed-point scale for loss sums
#define LOGCLAMP  (-100.0f)
#define NBIN0     2048
#define NBIN2     1024
#define TPB       256u
#define PK_CNT1   (1ull << 50)             // one element in the count field
#define PK_SUMMASK ((1ull << 50) - 1ull)

struct Ws {
  unsigned long long pos_cnt;   // number of positives
  unsigned long long pos_fix;   // fixed-point sum of positive losses
  unsigned long long k;         // negative_count (top-k size)
  unsigned long long cnt_sel;   // count of negatives strictly above current prefix
  unsigned long long sum_sel;   // fixed-point sum of those
  unsigned int b0;              // selected level-0 bin (key >> 21)
  unsigned int sel01;           // selected top-22 bits ((b0<<11)|b1)
  unsigned int pad[2];
  unsigned int       h0_cnt[NBIN0];
  unsigned long long h0_sum[NBIN0];
  unsigned int       h1_cnt[NBIN0];
  unsigned long long h1_sum[NBIN0];
  unsigned int       h2_cnt[NBIN2];
  unsigned long long h2_sum[NBIN2];
};

__device__ __forceinline__ unsigned long long fixp(float x) {
  // x >= 0 always (losses are non-negative)
  return (unsigned long long)(x * FSCALE + 0.5f);
}

// CDNA5 async global->LDS copy (ASYNCcnt-tracked), GVS addressing:
// mem = SGPR64 base + VGPR32 byte offset; vdst = per-lane LDS byte address.
#define ASYNC_LD_B128(ldsoff, saddr64, voff)                                   \
  asm volatile("global_load_async_to_lds_b128 %0, %1, %2"                      \
               :: "v"(ldsoff), "v"(voff), "s"(saddr64) : "memory")
#define WAIT_ASYNC(n) asm volatile("s_wait_asynccnt " #n ::: "memory")

// ---------------------------------------------------------------------------
// Pass 1: positive stats + level-0 histogram of negative-loss keys.
// Tiles of 1024 floats staged through LDS with double-buffered async loads.
// ---------------------------------------------------------------------------
__global__ __launch_bounds__(256) void k_pass1(const float* __restrict__ pred,
                                               const float* __restrict__ gt,
                                               unsigned N, Ws* __restrict__ ws)
{
  __shared__ unsigned long long hp[NBIN0];          // packed {cnt,sum}
  __shared__ float4 sp[2][TPB];
  __shared__ float4 sg[2][TPB];
  __shared__ unsigned long long s_pc, s_pf;

  const unsigned tid = threadIdx.x;
  for (unsigned i = tid; i < NBIN0; i += TPB) hp[i] = 0ull;
  if (tid == 0) { s_pc = 0ull; s_pf = 0ull; }
  __syncthreads();

  unsigned long long lpc = 0ull, lpf = 0ull;

  auto proc = [&](float p, float g) {
    if (g > 0.5f) {
      float v = -fmaxf(logf(p), LOGCLAMP);          // -clip(log(pred))
      lpc += 1ull;
      lpf += fixp(v);
    } else {
      float v = -fmaxf(log1pf(-p), LOGCLAMP);       // -clip(log1p(-pred))
      unsigned key = __float_as_uint(v);            // non-neg float: uint order
      atomicAdd(&hp[key >> 21], PK_CNT1 | fixp(v)); // single ds_add_u64
    }
  };

  const unsigned long long pbase = (unsigned long long)pred;
  const unsigned long long gbase = (unsigned long long)gt;
  const unsigned lds_p0 = (unsigned)(unsigned long long)&sp[0][tid];
  const unsigned lds_p1 = (unsigned)(unsigned long long)&sp[1][tid];
  const unsigned lds_g0 = (unsigned)(unsigned long long)&sg[0][tid];
  const unsigned lds_g1 = (unsigned)(unsigned long long)&sg[1][tid];

  const unsigned fullTiles = N / (TPB * 4u);        // 1024 floats per tile
  const unsigned stride = gridDim.x;
  unsigned t = blockIdx.x;
  unsigned buf = 0u;

  if (t < fullTiles) {                              // prologue: prime buffer 0
    const unsigned bo = (t * TPB + tid) * 16u;
    ASYNC_LD_B128(lds_p0, pbase, bo);
    ASYNC_LD_B128(lds_g0, gbase, bo);
  }
  for (; t < fullTiles; t += stride) {
    const unsigned nt = t + stride;
    if (nt < fullTiles) {                           // issue next tile -> buf^1
      const unsigned bo = (nt * TPB + tid) * 16u;
      const unsigned lp = buf ? lds_p0 : lds_p1;
      const unsigned lg = buf ? lds_g0 : lds_g1;
      ASYNC_LD_B128(lp, pbase, bo);
      ASYNC_LD_B128(lg, gbase, bo);
      WAIT_ASYNC(0x2);                              // drain current tile only
    } else {
      WAIT_ASYNC(0x0);
    }
    const float4 p4 = sp[buf][tid];                 // each lane reads own slot
    const float4 g4 = sg[buf][tid];
    proc(p4.x, g4.x); proc(p4.y, g4.y); proc(p4.z, g4.z); proc(p4.w, g4.w);
    buf ^= 1u;
  }

  // scalar tail (N not multiple of 1024)
  for (unsigned i = fullTiles * (TPB * 4u) + blockIdx.x * TPB + tid; i < N;
       i += stride * TPB)
    proc(pred[i], gt[i]);

  __syncthreads();
  if (lpc) { atomicAdd(&s_pc, lpc); atomicAdd(&s_pf, lpf); }
  __syncthreads();
  if (tid == 0 && s_pc) {
    atomicAdd(&ws->pos_cnt, s_pc);
    atomicAdd(&ws->pos_fix, s_pf);
  }
  for (unsigned i = tid; i < NBIN0; i += TPB) {
    const unsigned long long v = hp[i];
    if (v) {
      atomicAdd(&ws->h0_cnt[i], (unsigned)(v >> 50));
      atomicAdd(&ws->h0_sum[i], v & PK_SUMMASK);
    }
  }
}

// ---------------------------------------------------------------------------
// Pass 2: level-1 histogram (bits 20:10) among negatives whose level-0 bin
// matches b0. Inputs are L2-resident after pass 1 (105 MB < 192 MB L2).
// ---------------------------------------------------------------------------
__global__ __launch_bounds__(256) void k_pass2(const float* __restrict__ pred,
                                               const float* __restrict__ gt,
                                               unsigned N, Ws* __restrict__ ws)
{
  __shared__ unsigned long long hp[NBIN0];
  const unsigned tid = threadIdx.x;
  for (unsigned i = tid; i < NBIN0; i += TPB) hp[i] = 0ull;
  __syncthreads();

  const unsigned b0 = ws->b0;                       // 0xFFFFFFFF => match none
  const unsigned n4 = N / 4u;
  const unsigned gsz = gridDim.x * TPB;
  const float4* p4 = (const float4*)pred;
  const float4* g4 = (const float4*)gt;

  auto proc = [&](float p, float g) {
    if (g < 0.5f) {
      float v = -fmaxf(log1pf(-p), LOGCLAMP);
      unsigned key = __float_as_uint(v);
      if ((key >> 21) == b0)
        atomicAdd(&hp[(key >> 10) & 2047u], PK_CNT1 | fixp(v));
    }
  };

  for (unsigned i = blockIdx.x * TPB + tid; i < n4; i += gsz) {
    __builtin_prefetch(p4 + i + gsz, 0, 0);         // global_prefetch_b8
    __builtin_prefetch(g4 + i + gsz, 0, 0);
    const float4 p = p4[i];
    const float4 g = g4[i];
    proc(p.x, g.x); proc(p.y, g.y); proc(p.z, g.z); proc(p.w, g.w);
  }
  for (unsigned i = n4 * 4u + blockIdx.x * TPB + tid; i < N; i += gsz)
    proc(pred[i], gt[i]);

  __syncthreads();
  for (unsigned i = tid; i < NBIN0; i += TPB) {
    const unsigned long long v = hp[i];
    if (v) {
      atomicAdd(&ws->h1_cnt[i], (unsigned)(v >> 50));
      atomicAdd(&ws->h1_sum[i], v & PK_SUMMASK);
    }
  }
}

// ---------------------------------------------------------------------------
// Pass 3: level-2 histogram (bits 9:0) among negatives whose top-22 bits match.
// ---------------------------------------------------------------------------
__global__ __launch_bounds__(256) void k_pass3(const float* __restrict__ pred,
                                               const float* __restrict__ gt,
                                               unsigned N, Ws* __restrict__ ws)
{
  __shared__ unsigned long long hp[NBIN2];
  const unsigned tid = threadIdx.x;
  for (unsigned i = tid; i < NBIN2; i += TPB) hp[i] = 0ull;
  __syncthreads();

  const unsigned sel01 = ws->sel01;                 // 0xFFFFFFFF => match none
  const unsigned n4 = N / 4u;
  const unsigned gsz = gridDim.x * TPB;
  const float4* p4 = (const float4*)pred;
  const float4* g4 = (const float4*)gt;

  auto proc = [&](float p, float g) {
    if (g < 0.5f) {
      float v = -fmaxf(log1pf(-p), LOGCLAMP);
      unsigned key = __float_as_uint(v);
      if ((key >> 10) == sel01)
        atomicAdd(&hp[key & 1023u], PK_CNT1 | fixp(v));
    }
  };

  for (unsigned i = blockIdx.x * TPB + tid; i < n4; i += gsz) {
    __builtin_prefetch(p4 + i + gsz, 0, 0);
    __builtin_prefetch(g4 + i + gsz, 0, 0);
    const float4 p = p4[i];
    const float4 g = g4[i];
    proc(p.x, g.x); proc(p.y, g.y); proc(p.z, g.z); proc(p.w, g.w);
  }
  for (unsigned i = n4 * 4u + blockIdx.x * TPB + tid; i < N; i += gsz)
    proc(pred[i], gt[i]);

  __syncthreads();
  for (unsigned i = tid; i < NBIN2; i += TPB) {
    const unsigned long long v = hp[i];
    if (v) {
      atomicAdd(&ws->h2_cnt[i], (unsigned)(v >> 50));
      atomicAdd(&ws->h2_sum[i], v & PK_SUMMASK);
    }
  }
}

// ---------------------------------------------------------------------------
// Scans: locate the bin containing the k-th largest key at each level.
// Parallel 8-bin (4-bin at level 2) chunk totals, then a short serial walk over 256 LDS chunks.
// ---------------------------------------------------------------------------
__global__ void k_scan0(unsigned N, Ws* ws)
{
  __shared__ unsigned long long cc[TPB], cs[TPB];
  const unsigned tid = threadIdx.x;
  unsigned long long c = 0, s = 0;
  for (unsigned b = tid * 8u; b < tid * 8u + 8u; ++b) { c += ws->h0_cnt[b]; s += ws->h0_sum[b]; }
  cc[tid] = c; cs[tid] = s;
  __syncthreads();
  if (tid) return;

  const unsigned long long pos  = ws->pos_cnt;
  const unsigned long long negT = (unsigned long long)N - pos;
  const unsigned long long k    = pos < negT ? pos : negT;  // ratio=1, floor==pos
  ws->k = k;
  if (k == 0ull) {
    ws->b0 = 0xFFFFFFFFu; ws->sel01 = 0xFFFFFFFFu;
    ws->cnt_sel = 0ull; ws->sum_sel = 0ull;
    return;
  }
  unsigned long long cum = 0, sum = 0;
  int ch = 255;
  for (; ch > 0; --ch) { if (cum + cc[ch] >= k) break; cum += cc[ch]; sum += cs[ch]; }
  int b = ch * 8 + 7;
  for (; b > ch * 8; --b) {
    const unsigned long long x = ws->h0_cnt[b];
    if (cum + x >= k) break;
    cum += x; sum += ws->h0_sum[b];
  }
  ws->b0 = (unsigned)b;
  ws->cnt_sel = cum; ws->sum_sel = sum;
}

__global__ void k_scan1(Ws* ws)
{
  __shared__ unsigned long long cc[TPB], cs[TPB];
  const unsigned tid = threadIdx.x;
  unsigned long long c = 0, s = 0;
  for (unsigned b = tid * 8u; b < tid * 8u + 8u; ++b) { c += ws->h1_cnt[b]; s += ws->h1_sum[b]; }
  cc[tid] = c; cs[tid] = s;
  __syncthreads();
  if (tid) return;

  const unsigned long long k = ws->k;
  if (k == 0ull) { ws->sel01 = 0xFFFFFFFFu; return; }
  unsigned long long cum = ws->cnt_sel, sum = ws->sum_sel;
  int ch = 255;
  for (; ch > 0; --ch) { if (cum + cc[ch] >= k) break; cum += cc[ch]; sum += cs[ch]; }
  int b = ch * 8 + 7;
  for (; b > ch * 8; --b) {
    const unsigned long long x = ws->h1_cnt[b];
    if (cum + x >= k) break;
    cum += x; sum += ws->h1_sum[b];
  }
  ws->sel01 = (ws->b0 << 11) | (unsigned)b;
  ws->cnt_sel = cum; ws->sum_sel = sum;
}

__global__ void k_final(unsigned N, Ws* ws, float* out)
{
  __shared__ unsigned long long cc[TPB], cs[TPB];
  const unsigned tid = threadIdx.x;
  unsigned long long c = 0, s = 0;
  for (unsigned b = tid * 4u; b < tid * 4u + 4u; ++b) { c += ws->h2_cnt[b]; s += ws->h2_sum[b]; }
  cc[tid] = c; cs[tid] = s;
  __syncthreads();
  if (tid) return;

  const unsigned long long pos = ws->pos_cnt;
  const unsigned long long k   = ws->k;
  const double pos_sum = (double)ws->pos_fix * (1.0 / (double)FSCALE);
  double neg_sum = 0.0;
  if (k) {
    unsigned long long cum = ws->cnt_sel, sum = ws->sum_sel;
    int ch = 255;
    for (; ch > 0; --ch) { if (cum + cc[ch] >= k) break; cum += cc[ch]; sum += cs[ch]; }
    int b = ch * 4 + 3;
    for (; b > ch * 4; --b) {
      const unsigned long long x = ws->h2_cnt[b];
      if (cum + x >= k) break;
      cum += x; sum += ws->h2_sum[b];
    }
    // every element in bin b has the exact bit pattern T -> ties handled exactly
    const unsigned tbits = (ws->sel01 << 10) | (unsigned)b;
    const double T = (double)__uint_as_float(tbits);
    neg_sum = (double)sum * (1.0 / (double)FSCALE) + (double)(k - cum) * T;
  }
  const double denom = (double)(pos + k) + 1e-5;   // EPS
  out[0] = (float)((pos_sum + neg_sum) / denom);
}

__global__ void k_zero(unsigned* p, unsigned n)
{
  for (unsigned i = blockIdx.x * blockDim.x + threadIdx.x; i < n;
       i += gridDim.x * blockDim.x)
    p[i] = 0u;
}

// ---------------------------------------------------------------------------
extern "C" void kernel_launch(void* const* d_in, const int* in_sizes, int n_in,
                              void* d_out, int out_size, void* d_ws, size_t ws_size,
                              hipStream_t stream)
{
  (void)n_in; (void)out_size; (void)ws_size;
  const float* pred = (const float*)d_in[0];
  const float* gt   = (const float*)d_in[1];
  const unsigned N  = (unsigned)in_sizes[0];
  Ws* ws            = (Ws*)d_ws;
  float* out        = (float*)d_out;

  const unsigned words = (unsigned)((sizeof(Ws) + 3) / 4);
  k_zero <<<64, TPB, 0, stream>>>((unsigned*)d_ws, words);
  k_pass1<<<2048, TPB, 0, stream>>>(pred, gt, N, ws);
  k_scan0<<<1, TPB, 0, stream>>>(N, ws);
  k_pass2<<<2048, TPB, 0, stream>>>(pred, gt, N, ws);
  k_scan1<<<1, TPB, 0, stream>>>(ws);
  k_pass3<<<2048, TPB, 0, stream>>>(pred, gt, N, ws);
  k_final<<<1, TPB, 0, stream>>>(N, ws, out);
}